// FrequencyAttention2_2250562863501
// MI455X (gfx1250) — compile-verified
//
#include <hip/hip_runtime.h>
#include <math.h>

// Problem constants (from reference)
#define BB 32
#define LL 512
#define DD 512
#define TOPK 4
#define FF 257              // rfft bins = L/2 + 1
#define FP 272              // FF padded up to multiple of 16 for WMMA tiling

typedef float v2f __attribute__((ext_vector_type(2)));
typedef float v8f __attribute__((ext_vector_type(8)));

// ---------------------------------------------------------------------------
// Transpose a 512x512 matrix (for W.T in the projections)
// ---------------------------------------------------------------------------
__global__ __launch_bounds__(256) void transpose512(const float* __restrict__ in,
                                                    float* __restrict__ out) {
  __shared__ float tile[32][33];
  int bx = blockIdx.x * 32, by = blockIdx.y * 32;
  int tx = threadIdx.x, ty = threadIdx.y;  // block (32, 8)
#pragma unroll
  for (int i = 0; i < 32; i += 8)
    tile[ty + i][tx] = in[(size_t)(by + ty + i) * DD + (bx + tx)];
  __syncthreads();
#pragma unroll
  for (int i = 0; i < 32; i += 8)
    out[(size_t)(bx + ty + i) * DD + (by + tx)] = tile[tx][ty + i];
}

// ---------------------------------------------------------------------------
// Build DFT matrices.
//  Forward (rfft):  Fre[f][l] =  cos(2*pi*f*l/L),  Fim[f][l] = -sin(2*pi*f*l/L)
//  Inverse (irfft): IC[l][f]  =  w_f*cos(2*pi*f*l/L)/L, IS[l][f] = -w_f*sin(...)/L
//  with w_f = 1 for f in {0, L/2} else 2 (Hermitian folding). Rows/cols with
//  f >= FF are zero so the FP padding is inert through the whole pipeline.
// ---------------------------------------------------------------------------
__global__ void init_dft(float* __restrict__ Fre, float* __restrict__ Fim,
                         float* __restrict__ IC, float* __restrict__ IS) {
  int i = blockIdx.x * blockDim.x + threadIdx.x;
  if (i >= FP * LL) return;
  const float w0 = 6.28318530717958647692f / (float)LL;
  // forward, layout [FP][LL]
  {
    int f = i / LL, l = i % LL;
    int p = (f * l) & (LL - 1);  // exact phase reduction mod L
    float s, c;
    __sincosf(w0 * (float)p, &s, &c);
    bool valid = (f < FF);
    Fre[i] = valid ? c : 0.f;
    Fim[i] = valid ? -s : 0.f;
  }
  // inverse, layout [LL][FP]
  {
    int l = i / FP, f = i % FP;
    int p = (f * l) & (LL - 1);
    float s, c;
    __sincosf(w0 * (float)p, &s, &c);
    float w = (f == 0 || f == FF - 1) ? 1.f : 2.f;
    bool valid = (f < FF);
    IC[i] = valid ? (w * c / (float)LL) : 0.f;
    IS[i] = valid ? (-w * s / (float)LL) : 0.f;
  }
}

// ---------------------------------------------------------------------------
// Generic batched fp32 WMMA GEMM:  C[b] (MxN) = A (MxKd) @ B[b] (KdxN) [+bias]
// A is shared across batches. N multiple of 64; Kd multiple of 16; M guarded.
// Block = 256 threads (8 waves); tile 128x64, K-step 16, LDS double-buffered.
// Each wave owns a 32x32 slab = 4 v_wmma_f32_16x16x4_f32 accumulators
// (16 WMMA per K-tile). Fragment K-pairs are contiguous 8B-aligned in LDS
// (A kept [m][k], B stored transposed [n][k], pitch 18 floats) so each WMMA
// operand pair is filled by a single ds_load_b64 — no register marshalling.
// ---------------------------------------------------------------------------
__global__ __launch_bounds__(256) void wmma_gemm_f32(
    const float* __restrict__ A, const float* __restrict__ Bm,
    float* __restrict__ C, const float* __restrict__ bias,
    int M, int N, int Kd, long batchStrideB, long batchStrideC, int accumulate) {
  __shared__ __align__(16) float As[2][128][18];   // [buf][m][k], 8B-aligned pairs
  __shared__ __align__(16) float Bst[2][64][18];   // [buf][n][k] (transposed)

  const float* Bb = Bm + (size_t)blockIdx.z * (size_t)batchStrideB;
  float* Cb = C + (size_t)blockIdx.z * (size_t)batchStrideC;

  const int t = threadIdx.x;
  const int lane = t & 31;
  const int wave = t >> 5;   // 0..7
  const int wm = wave & 3;   // 4 sub-tiles of 32 rows
  const int wn = wave >> 2;  // 2 sub-tiles of 32 cols
  const int mBase = blockIdx.y * 128;
  const int nBase = blockIdx.x * 64;
  const int mOff = wm * 32;
  const int nOff = wn * 32;
  const int hi = (lane >> 4) & 1;  // lanes 16-31 hold the K=2,3 pair
  const int ml = lane & 15;

  v8f acc00 = {}, acc01 = {}, acc10 = {}, acc11 = {};

  // ---- prologue: stage K-tile 0 into buffer 0 ----
#pragma unroll
  for (int i = 0; i < 8; ++i) {
    int idx = t + i * 256;
    int r = idx >> 4, c = idx & 15;
    int gr = mBase + r;
    As[0][r][c] = (gr < M) ? A[(size_t)gr * Kd + c] : 0.f;
  }
#pragma unroll
  for (int i = 0; i < 4; ++i) {
    int idx = t + i * 256;
    int r = idx >> 6, c = idx & 63;
    Bst[0][c][r] = Bb[(size_t)r * N + (nBase + c)];
  }
  __syncthreads();

  int buf = 0;
  for (int kb = 0; kb < Kd; kb += 16) {
    const int kbN = kb + 16;
    const bool haveNext = (kbN < Kd);
    float ra[8];
    float rb[4];
    if (haveNext) {
      // issue global loads for the next K-tile (overlap with WMMA below)
#pragma unroll
      for (int i = 0; i < 8; ++i) {
        int idx = t + i * 256;
        int r = idx >> 4, c = idx & 15;
        int gr = mBase + r;
        ra[i] = (gr < M) ? A[(size_t)gr * Kd + (kbN + c)] : 0.f;
      }
#pragma unroll
      for (int i = 0; i < 4; ++i) {
        int idx = t + i * 256;
        int r = idx >> 6, c = idx & 63;
        rb[i] = Bb[(size_t)(kbN + r) * N + (nBase + c)];
      }
      // warm L2 for the tile after next (global_prefetch_b8)
      if (kbN + 16 < Kd) {
        int gr = mBase + (t >> 4);
        if (gr < M) __builtin_prefetch(&A[(size_t)gr * Kd + (kbN + 16)], 0, 0);
        __builtin_prefetch(&Bb[(size_t)(kbN + 16 + (t >> 6)) * N + nBase + (t & 63)], 0, 0);
      }
    }

    // ---- 16 WMMA ops on the current buffer ----
#pragma unroll
    for (int kk = 0; kk < 16; kk += 4) {
      const int k2 = kk + 2 * hi;
      v2f a0 = *(const v2f*)&As[buf][mOff + ml][k2];
      v2f a1 = *(const v2f*)&As[buf][mOff + 16 + ml][k2];
      v2f b0 = *(const v2f*)&Bst[buf][nOff + ml][k2];
      v2f b1 = *(const v2f*)&Bst[buf][nOff + 16 + ml][k2];
      acc00 = __builtin_amdgcn_wmma_f32_16x16x4_f32(false, a0, false, b0,
                                                    (short)0, acc00, false, false);
      acc01 = __builtin_amdgcn_wmma_f32_16x16x4_f32(false, a0, false, b1,
                                                    (short)0, acc01, false, false);
      acc10 = __builtin_amdgcn_wmma_f32_16x16x4_f32(false, a1, false, b0,
                                                    (short)0, acc10, false, false);
      acc11 = __builtin_amdgcn_wmma_f32_16x16x4_f32(false, a1, false, b1,
                                                    (short)0, acc11, false, false);
    }

    if (haveNext) {
      const int nb = buf ^ 1;
#pragma unroll
      for (int i = 0; i < 8; ++i) {
        int idx = t + i * 256;
        As[nb][idx >> 4][idx & 15] = ra[i];
      }
#pragma unroll
      for (int i = 0; i < 4; ++i) {
        int idx = t + i * 256;
        Bst[nb][idx & 63][idx >> 6] = rb[i];
      }
    }
    __syncthreads();
    buf ^= 1;
  }

  // C layout: VGPR r -> row +r (lanes 0-15) / row +8+r (lanes 16-31)
#pragma unroll
  for (int r = 0; r < 8; ++r) {
    const int row0 = mBase + mOff + r + (hi ? 8 : 0);
    const int row1 = row0 + 16;
    const int c0 = nBase + nOff + ml;
    const int c1 = c0 + 16;
    if (row0 < M) {
      float v0 = acc00[r], v1 = acc01[r];
      if (bias) { v0 += bias[c0]; v1 += bias[c1]; }
      size_t o0 = (size_t)row0 * N + c0;
      size_t o1 = (size_t)row0 * N + c1;
      if (accumulate) { v0 += Cb[o0]; v1 += Cb[o1]; }
      Cb[o0] = v0;
      Cb[o1] = v1;
    }
    if (row1 < M) {
      float v0 = acc10[r], v1 = acc11[r];
      if (bias) { v0 += bias[c0]; v1 += bias[c1]; }
      size_t o0 = (size_t)row1 * N + c0;
      size_t o1 = (size_t)row1 * N + c1;
      if (accumulate) { v0 += Cb[o0]; v1 += Cb[o1]; }
      Cb[o0] = v0;
      Cb[o1] = v1;
    }
  }
}

// ---------------------------------------------------------------------------
// Top-4 amplitude mask along the frequency axis, per (b, d).
// Keep entries with |x|^2 >= (4th largest |x|^2); zero the rest.
// ---------------------------------------------------------------------------
__global__ void topk_mask(float* __restrict__ re, float* __restrict__ im) {
  int idx = blockIdx.x * blockDim.x + threadIdx.x;
  if (idx >= BB * DD) return;
  int b = idx / DD, d = idx % DD;
  float* pr = re + (size_t)b * FP * DD + d;
  float* pi = im + (size_t)b * FP * DD + d;
  float t0 = -1.f, t1 = -1.f, t2 = -1.f, t3 = -1.f;  // top-4 descending
  for (int f = 0; f < FF; ++f) {
    size_t o = (size_t)f * DD;
    float r = pr[o], i2 = pi[o];
    float a = r * r + i2 * i2;
    if (a > t3) {
      if (a > t0)      { t3 = t2; t2 = t1; t1 = t0; t0 = a; }
      else if (a > t1) { t3 = t2; t2 = t1; t1 = a; }
      else if (a > t2) { t3 = t2; t2 = a; }
      else             { t3 = a; }
    }
  }
  float thr = t3;
  for (int f = 0; f < FF; ++f) {
    size_t o = (size_t)f * DD;
    float r = pr[o], i2 = pi[o];
    if (r * r + i2 * i2 < thr) { pr[o] = 0.f; pi[o] = 0.f; }
  }
}

// att = qf * conj(kf), in place into the qf planes
__global__ void attmul(float* __restrict__ qr, float* __restrict__ qi,
                       const float* __restrict__ kr, const float* __restrict__ ki,
                       size_t n) {
  size_t i = (size_t)blockIdx.x * blockDim.x + threadIdx.x;
  if (i >= n) return;
  float ar = qr[i], ai = qi[i], br = kr[i], bi = ki[i];
  qr[i] = ar * br + ai * bi;
  qi[i] = ai * br - ar * bi;
}

// Row softmax over D=512 (one block of 256 threads per row, 2 elems/thread)
__global__ __launch_bounds__(256) void softmax_row(float* __restrict__ x) {
  __shared__ float red[256];
  float* p = x + (size_t)blockIdx.x * DD;
  int t = threadIdx.x;
  float v0 = p[t], v1 = p[t + 256];
  red[t] = fmaxf(v0, v1);
  __syncthreads();
  for (int s = 128; s > 0; s >>= 1) {
    if (t < s) red[t] = fmaxf(red[t], red[t + s]);
    __syncthreads();
  }
  float m = red[0];
  __syncthreads();
  float e0 = __expf(v0 - m), e1 = __expf(v1 - m);
  red[t] = e0 + e1;
  __syncthreads();
  for (int s = 128; s > 0; s >>= 1) {
    if (t < s) red[t] += red[t + s];
    __syncthreads();
  }
  float inv = 1.f / red[0];
  p[t] = e0 * inv;
  p[t + 256] = e1 * inv;
}

// ctx = att * vf (complex), written into the freed kf planes
__global__ void ctxmul(float* __restrict__ cr, float* __restrict__ ci,
                       const float* __restrict__ ar, const float* __restrict__ ai,
                       const float* __restrict__ vr, const float* __restrict__ vi,
                       size_t n) {
  size_t i = (size_t)blockIdx.x * blockDim.x + threadIdx.x;
  if (i >= n) return;
  float x = ar[i] * vr[i] - ai[i] * vi[i];
  float y = ar[i] * vi[i] + ai[i] * vr[i];
  cr[i] = x;
  ci[i] = y;
}

// out = LayerNorm(query + irfft_out) * gamma + beta
__global__ __launch_bounds__(256) void residual_ln(
    const float* __restrict__ query, const float* __restrict__ add,
    const float* __restrict__ gamma, const float* __restrict__ beta,
    float* __restrict__ out) {
  __shared__ float red[256];
  size_t row = blockIdx.x;
  const float* q = query + row * DD;
  const float* a = add + row * DD;
  int t = threadIdx.x;
  float x0 = q[t] + a[t];
  float x1 = q[t + 256] + a[t + 256];
  red[t] = x0 + x1;
  __syncthreads();
  for (int s = 128; s > 0; s >>= 1) {
    if (t < s) red[t] += red[t + s];
    __syncthreads();
  }
  float mu = red[0] * (1.f / (float)DD);
  __syncthreads();
  float d0 = x0 - mu, d1 = x1 - mu;
  red[t] = d0 * d0 + d1 * d1;
  __syncthreads();
  for (int s = 128; s > 0; s >>= 1) {
    if (t < s) red[t] += red[t + s];
    __syncthreads();
  }
  float inv = rsqrtf(red[0] * (1.f / (float)DD) + 1e-5f);
  out[row * DD + t] = d0 * inv * gamma[t] + beta[t];
  out[row * DD + t + 256] = d1 * inv * gamma[t + 256] + beta[t + 256];
}

// ---------------------------------------------------------------------------
// Orchestration. Workspace demand: ~53.25M floats (~203 MB).
// ---------------------------------------------------------------------------
extern "C" void kernel_launch(void* const* d_in, const int* in_sizes, int n_in,
                              void* d_out, int out_size, void* d_ws, size_t ws_size,
                              hipStream_t stream) {
  (void)in_sizes; (void)n_in; (void)out_size; (void)ws_size;
  const float* key   = (const float*)d_in[0];
  const float* value = (const float*)d_in[1];
  const float* query = (const float*)d_in[2];
  const float* Wk    = (const float*)d_in[3];
  const float* bk    = (const float*)d_in[4];
  const float* Wv    = (const float*)d_in[5];
  const float* bv    = (const float*)d_in[6];
  const float* Wq    = (const float*)d_in[7];
  const float* bq    = (const float*)d_in[8];
  const float* gamma = (const float*)d_in[9];
  const float* beta  = (const float*)d_in[10];
  float* out = (float*)d_out;

  float* ws = (float*)d_ws;
  size_t off = 0;
  auto alloc = [&](size_t n) { float* p = ws + off; off += n; return p; };

  const size_t DM = (size_t)DD * DD;        // 512*512
  const size_t FWD = (size_t)FP * LL;       // DFT matrix elems
  const size_t TD = (size_t)BB * LL * DD;   // time-domain tensor
  const size_t FD = (size_t)BB * FP * DD;   // freq-domain plane

  float* WkT = alloc(DM);
  float* WvT = alloc(DM);
  float* WqT = alloc(DM);
  float* Fre = alloc(FWD);
  float* Fim = alloc(FWD);
  float* IC  = alloc(FWD);
  float* IS  = alloc(FWD);
  float* kbuf = alloc(TD);   // k projection; later reused as irfft output
  float* vbuf = alloc(TD);
  float* qbuf = alloc(TD);
  float* kfr = alloc(FD);    // later reused as ctx_re
  float* kfi = alloc(FD);    // later reused as ctx_im
  float* vfr = alloc(FD);
  float* vfi = alloc(FD);
  float* qfr = alloc(FD);    // later reused as att_re
  float* qfi = alloc(FD);    // later reused as att_im

  // 1) W.T for the projections
  dim3 tb(32, 8), tg(16, 16);
  transpose512<<<tg, tb, 0, stream>>>(Wk, WkT);
  transpose512<<<tg, tb, 0, stream>>>(Wv, WvT);
  transpose512<<<tg, tb, 0, stream>>>(Wq, WqT);

  // 2) DFT matrices
  {
    int n = FP * LL;
    init_dft<<<(n + 255) / 256, 256, 0, stream>>>(Fre, Fim, IC, IS);
  }

  // 3) Projections: [16384x512] @ [512x512] + bias  (WMMA fp32)
  {
    dim3 g(DD / 64, (BB * LL) / 128, 1);  // (8, 128, 1)
    wmma_gemm_f32<<<g, 256, 0, stream>>>(key,   WkT, kbuf, bk, BB * LL, DD, DD, 0, 0, 0);
    wmma_gemm_f32<<<g, 256, 0, stream>>>(value, WvT, vbuf, bv, BB * LL, DD, DD, 0, 0, 0);
    wmma_gemm_f32<<<g, 256, 0, stream>>>(query, WqT, qbuf, bq, BB * LL, DD, DD, 0, 0, 0);
  }

  // 4) rfft as batched GEMM: [FPx512] @ [512x512] per batch  (WMMA fp32)
  {
    dim3 g(DD / 64, (FP + 127) / 128, BB);  // (8, 3, 32)
    long sB = (long)LL * DD, sC = (long)FP * DD;
    wmma_gemm_f32<<<g, 256, 0, stream>>>(Fre, kbuf, kfr, nullptr, FP, DD, LL, sB, sC, 0);
    wmma_gemm_f32<<<g, 256, 0, stream>>>(Fim, kbuf, kfi, nullptr, FP, DD, LL, sB, sC, 0);
    wmma_gemm_f32<<<g, 256, 0, stream>>>(Fre, vbuf, vfr, nullptr, FP, DD, LL, sB, sC, 0);
    wmma_gemm_f32<<<g, 256, 0, stream>>>(Fim, vbuf, vfi, nullptr, FP, DD, LL, sB, sC, 0);
    wmma_gemm_f32<<<g, 256, 0, stream>>>(Fre, qbuf, qfr, nullptr, FP, DD, LL, sB, sC, 0);
    wmma_gemm_f32<<<g, 256, 0, stream>>>(Fim, qbuf, qfi, nullptr, FP, DD, LL, sB, sC, 0);
  }

  // 5) top-4 frequency mask per (b, d)
  {
    int n = BB * DD;  // 16384 threads
    topk_mask<<<(n + 255) / 256, 256, 0, stream>>>(kfr, kfi);
    topk_mask<<<(n + 255) / 256, 256, 0, stream>>>(vfr, vfi);
    topk_mask<<<(n + 255) / 256, 256, 0, stream>>>(qfr, qfi);
  }

  // 6) att = qf * conj(kf)   (in place into qf planes)
  {
    size_t n = FD;
    attmul<<<(unsigned)((n + 255) / 256), 256, 0, stream>>>(qfr, qfi, kfr, kfi, n);
  }

  // 7) softmax over D of real and imag parts
  {
    unsigned rows = BB * FP;  // 8704 rows (padded rows are inert zeros)
    softmax_row<<<rows, 256, 0, stream>>>(qfr);
    softmax_row<<<rows, 256, 0, stream>>>(qfi);
  }

  // 8) ctx = att * vf  (complex), into the freed kf planes
  {
    size_t n = FD;
    ctxmul<<<(unsigned)((n + 255) / 256), 256, 0, stream>>>(kfr, kfi, qfr, qfi, vfr, vfi, n);
  }

  // 9) irfft as batched GEMM: out = IC@ctx_re + IS@ctx_im  (WMMA fp32, acc)
  {
    dim3 g(DD / 64, LL / 128, BB);  // (8, 4, 32)
    long sB = (long)FP * DD, sC = (long)LL * DD;
    wmma_gemm_f32<<<g, 256, 0, stream>>>(IC, kfr, kbuf, nullptr, LL, DD, FP, sB, sC, 0);
    wmma_gemm_f32<<<g, 256, 0, stream>>>(IS, kfi, kbuf, nullptr, LL, DD, FP, sB, sC, 1);
  }

  // 10) residual + layernorm
  {
    unsigned rows = BB * LL;  // 16384
    residual_ln<<<rows, 256, 0, stream>>>(query, kbuf, gamma, beta, out);
  }
}